// TtLlamaAttention_62758062129153
// MI455X (gfx1250) — compile-verified
//
#include <hip/hip_runtime.h>
#include <hip/hip_bf16.h>

// ---------------------------------------------------------------------------
// Llama GQA decode attention, fp32, MI455X (gfx1250, wave32).
// Memory-bound (~0.70 GB streamed -> ~30us @ 23.3 TB/s). All GEMMs use
// v_wmma_f32_16x16x4_f32. SDPA is split-S flash-decode: K/V tiles stream
// through double-buffered LDS via global_load_async_to_lds_b128, pipelined
// with s_wait_asynccnt (in-order async completion), Q held in registers.
// ---------------------------------------------------------------------------

typedef float v2f __attribute__((ext_vector_type(2)));
typedef float v8f __attribute__((ext_vector_type(8)));

#define B_      4
#define D_      4096
#define HQ_     32
#define HKV_    8
#define DH_     128
#define S_      16384
#define G_      4            // HQ/HKV
#define QKV_    6144         // (HQ + 2*HKV) * DH
#define CHUNK_  512
#define NCHUNK_ 32           // S / CHUNK
#define TILES_  32           // CHUNK / 16
#define LDP     132          // DH + 4 pad (16B-aligned rows, bank-conflict free)
#define SPP     17           // P-tile row pitch

// D = A(16x4) * B(4x16) + C, fp32, wave32.  Layouts per CDNA5 ISA 7.12.2:
//  A: lane l -> M = l&15, VGPR{0,1} = K = 2*(l>>4) + {0,1}
//  B: lane l -> N = l&15, VGPR{0,1} = K = 2*(l>>4) + {0,1}
//  C/D: lane l -> N = l&15, VGPR r -> M = r + 8*(l>>4)
__device__ __forceinline__ v8f wmma4(v2f a, v2f b, v8f c) {
  return __builtin_amdgcn_wmma_f32_16x16x4_f32(
      false, a, false, b, (short)0, c, false, false);
}

// Issue async DMA of one 16x128 fp32 tile pair (K,V) into LDS buffers.
// 32 in-order async ops (16 iters x 2), 16B per lane per op.
__device__ __forceinline__ void stage_async(const float* gk, const float* gv,
                                            float* bK, float* bV, int lane) {
  for (int i = lane; i < 512; i += 32) {
    const int row = i >> 5, c4 = i & 31;
    const float* pk = gk + (size_t)row * DH_ + c4 * 4;
    const float* pv = gv + (size_t)row * DH_ + c4 * 4;
    const unsigned lk = (unsigned)(uintptr_t)&bK[row * LDP + c4 * 4];
    const unsigned lv = (unsigned)(uintptr_t)&bV[row * LDP + c4 * 4];
    asm volatile("global_load_async_to_lds_b128 %0, %1, off"
                 :: "v"(lk), "v"(pk) : "memory");
    asm volatile("global_load_async_to_lds_b128 %0, %1, off"
                 :: "v"(lv), "v"(pv) : "memory");
  }
}

// ---------------------------------------------------------------------------
// Kernel 1: qkv = x(4x4096) @ wqkv(4096x6144).
// One wave computes two interleaved 16-col tiles (even/odd cols of a 32-col
// span): each weight row is one b64 load feeding two WMMAs. Lanes 4..15
// duplicate x rows 0..3 (no EXEC masking); duplicate rows never stored.
// ---------------------------------------------------------------------------
__global__ void k_qkv(const float* __restrict__ x, const float* __restrict__ wqkv,
                      float* __restrict__ qkv) {
  const int lane = threadIdx.x;
  const int nrow = lane & 15, hi = lane >> 4;
  const int n0 = blockIdx.x * 32;
  const int mrow = nrow & 3;
  const float2* x2 = reinterpret_cast<const float2*>(x);
  const float2* w2 = reinterpret_cast<const float2*>(wqkv);
  v8f acc0 = {}, acc1 = {};
  for (int k0 = 0; k0 < D_; k0 += 4) {
    const int ka = k0 + 2 * hi;                               // even
    const float2 av = x2[((size_t)mrow * D_ + ka) >> 1];
    const float2 w0 = w2[((size_t)ka * QKV_ + n0 + 2 * nrow) >> 1];
    const float2 w1 = w2[((size_t)(ka + 1) * QKV_ + n0 + 2 * nrow) >> 1];
    v2f a;  a.x = av.x;  a.y = av.y;
    v2f b0; b0.x = w0.x; b0.y = w1.x;                         // even cols
    v2f b1; b1.x = w0.y; b1.y = w1.y;                         // odd cols
    acc0 = wmma4(a, b0, acc0);
    acc1 = wmma4(a, b1, acc1);
  }
#pragma unroll
  for (int r = 0; r < 8; ++r) {
    const int m = r + 8 * hi;
    if (m < B_) {
      float2 st; st.x = acc0[r]; st.y = acc1[r];
      *reinterpret_cast<float2*>(&qkv[(size_t)m * QKV_ + n0 + 2 * nrow]) = st;
    }
  }
}

// ---------------------------------------------------------------------------
// Kernel 2: rotary. qrot = Q(32x128)@R, krot = K(8x128)@R per batch.
// blockIdx.x = b*24 + mt*8 + nt;  mt 0,1 -> q row-tiles, mt 2 -> k rows
// (lanes 8..15 duplicate k rows 0..7; duplicates not stored).
// ---------------------------------------------------------------------------
__global__ void k_rope(const float* __restrict__ qkv, const float* __restrict__ rot,
                       float* __restrict__ qrot, float* __restrict__ krot) {
  const int lane = threadIdx.x;
  const int nrow = lane & 15, hi = lane >> 4;
  const int idx = blockIdx.x;
  const int nt = idx & 7;
  const int mt = (idx >> 3) % 3;   // uniform per block
  const int b = idx / 24;
  const float* R = rot + (size_t)b * DH_ * DH_;
  const float* Abase = (mt < 2)
      ? qkv + (size_t)b * QKV_ + (size_t)(mt * 16 + nrow) * DH_
      : qkv + (size_t)b * QKV_ + HQ_ * DH_ + (size_t)(nrow & 7) * DH_;
  const float2* A2 = reinterpret_cast<const float2*>(Abase);
  v8f acc = {};
  for (int k0 = 0; k0 < DH_; k0 += 4) {
    const int ka = k0 + 2 * hi;                               // even
    const float2 av = A2[ka >> 1];
    v2f a, bb;
    a.x = av.x; a.y = av.y;
    bb.x = R[(size_t)ka * DH_ + nt * 16 + nrow];
    bb.y = R[(size_t)(ka + 1) * DH_ + nt * 16 + nrow];
    acc = wmma4(a, bb, acc);
  }
#pragma unroll
  for (int r = 0; r < 8; ++r) {
    const int m = r + 8 * hi;
    if (mt < 2)
      qrot[(size_t)b * HQ_ * DH_ + (size_t)(mt * 16 + m) * DH_ + nt * 16 + nrow] = acc[r];
    else if (m < HKV_)
      krot[(size_t)b * HKV_ * DH_ + (size_t)m * DH_ + nt * 16 + nrow] = acc[r];
  }
}

// ---------------------------------------------------------------------------
// Kernel 3: flash-decode partials. Grid (NCHUNK, HKV, B), 1 wave each.
// Fully-cached chunks: double-buffered async DMA pipeline (issue t+1, wait
// asynccnt<=32 for tile t, compute). Boundary chunks: serial per-tile path.
// Q lives in registers (rows duplicated mod 4; garbage rows never consumed).
// ---------------------------------------------------------------------------
__global__ void k_sdpa(const float* __restrict__ qrot, const float* __restrict__ krot,
                       const float* __restrict__ qkv, const float* __restrict__ cache_k,
                       const float* __restrict__ cache_v, const int* __restrict__ cur_pos,
                       float* __restrict__ part) {
  __shared__ float sK[2][16 * LDP];
  __shared__ float sV[2][16 * LDP];
  __shared__ float sP[16 * SPP];

  const int lane = threadIdx.x;
  const int nrow = lane & 15, hi = lane >> 4;
  const int chunk = blockIdx.x, h = blockIdx.y, b = blockIdx.z;
  const int cur = cur_pos[b];
  const int cbase = chunk * CHUNK_;
  const float scale = 0.088388347648318447f;  // 1/sqrt(128)

  float* P = part + (((size_t)(b * HKV_ + h)) * NCHUNK_ + chunk) * (G_ * DH_ + 2 * G_);

  // Fully-masked chunk: emit neutral partial and leave.
  if (cbase > cur) {
    for (int i = lane; i < G_ * DH_; i += 32) P[i] = 0.f;
    if (lane < G_) {
      P[G_ * DH_ + lane] = -1e30f;
      P[G_ * DH_ + G_ + lane] = 0.f;
    }
    return;
  }

  // Q (4 rows x 128) in registers, A-layout, rows duplicated mod 4.
  v2f aq[32];
  {
    const float2* q2 = reinterpret_cast<const float2*>(
        qrot + ((size_t)b * HQ_ + h * G_ + (nrow & 3)) * DH_);
#pragma unroll
    for (int j = 0; j < 32; ++j) {
      const float2 t = q2[2 * j + hi];
      aq[j].x = t.x; aq[j].y = t.y;
    }
  }

  float mr[G_] = {-1e30f, -1e30f, -1e30f, -1e30f};
  float lr[G_] = {0.f, 0.f, 0.f, 0.f};
  v8f acc[8];
#pragma unroll
  for (int nt = 0; nt < 8; ++nt) { v8f z = {}; acc[nt] = z; }

  // --- compute one 16-position tile resident in (bK, bV) ---
  auto compute_tile = [&](const float* bK, const float* bV, int pbase) {
    // Scores: S(16x16) = Q @ K_tile^T, 32 WMMA steps of K=4.
    v8f sc = {};
#pragma unroll
    for (int j = 0; j < 32; ++j) {
      const int ka = 4 * j + 2 * hi;
      v2f bb;
      bb.x = bK[nrow * LDP + ka]; bb.y = bK[nrow * LDP + ka + 1];  // B[k][n]=K[n][k]
      sc = wmma4(aq[j], bb, sc);
    }
    const bool dead = (pbase + nrow) > cur;
#pragma unroll
    for (int r = 0; r < 8; ++r) sc[r] = dead ? -1e9f : sc[r] * scale;

    // Online softmax, rows 0..3 (lanes 0..15 hold the 16 columns).
    float pf[G_];
#pragma unroll
    for (int g = 0; g < G_; ++g) {
      float v = sc[g];
      v = fmaxf(v, __shfl_xor(v, 1));
      v = fmaxf(v, __shfl_xor(v, 2));
      v = fmaxf(v, __shfl_xor(v, 4));
      v = fmaxf(v, __shfl_xor(v, 8));
      const float nm = fmaxf(mr[g], v);
      pf[g] = __expf(mr[g] - nm);
      mr[g] = nm;
    }
    float pv[8];
#pragma unroll
    for (int r = 0; r < 8; ++r) {
      const int row = r + 8 * hi;
      pv[r] = (row < G_) ? __expf(sc[r] - mr[row]) : 0.f;
    }
#pragma unroll
    for (int g = 0; g < G_; ++g) {
      float v = pv[g];
      v += __shfl_xor(v, 1);
      v += __shfl_xor(v, 2);
      v += __shfl_xor(v, 4);
      v += __shfl_xor(v, 8);
      lr[g] = lr[g] * pf[g] + v;
    }
    // P tile through LDS (C-layout -> A-layout), rescale accumulators.
#pragma unroll
    for (int r = 0; r < 8; ++r) sP[(r + 8 * hi) * SPP + nrow] = pv[r];
#pragma unroll
    for (int nt = 0; nt < 8; ++nt)
#pragma unroll
      for (int g = 0; g < G_; ++g) acc[nt][g] *= pf[g];
    // out(16x128) += P(16x16) @ V(16x128): 8 n-tiles x 4 WMMA steps.
#pragma unroll
    for (int nt = 0; nt < 8; ++nt) {
      v8f c = acc[nt];
#pragma unroll
      for (int k0 = 0; k0 < 16; k0 += 4) {
        const int ka = k0 + 2 * hi;
        v2f a, bb;
        a.x = sP[nrow * SPP + ka];            a.y = sP[nrow * SPP + ka + 1];
        bb.x = bV[ka * LDP + nt * 16 + nrow]; bb.y = bV[(ka + 1) * LDP + nt * 16 + nrow];
        c = wmma4(a, bb, c);
      }
      acc[nt] = c;
    }
  };

  const float* gK = cache_k + ((size_t)(b * HKV_ + h) * S_ + cbase) * DH_;
  const float* gV = cache_v + ((size_t)(b * HKV_ + h) * S_ + cbase) * DH_;

  if (cbase + CHUNK_ <= cur) {
    // ---- fully-cached chunk: double-buffered async pipeline ----
    stage_async(gK, gV, sK[0], sV[0], lane);
    for (int t = 0; t < TILES_; ++t) {
      if (t + 1 < TILES_) {
        stage_async(gK + (size_t)(t + 1) * 16 * DH_, gV + (size_t)(t + 1) * 16 * DH_,
                    sK[(t + 1) & 1], sV[(t + 1) & 1], lane);
        asm volatile("s_wait_asynccnt 0x20" ::: "memory");  // tile t landed
      } else {
        asm volatile("s_wait_asynccnt 0x0" ::: "memory");
      }
      compute_tile(sK[t & 1], sV[t & 1], cbase + t * 16);
    }
  } else {
    // ---- boundary chunk: serial tiles, frontier handled in registers ----
    const float4* K4  = reinterpret_cast<const float4*>(cache_k);
    const float4* V4  = reinterpret_cast<const float4*>(cache_v);
    const float4* Kn4 = reinterpret_cast<const float4*>(krot);
    const float4* Qk4 = reinterpret_cast<const float4*>(qkv);
    const size_t knbase4 = ((size_t)(b * HKV_ + h) * DH_) >> 2;
    const size_t vnbase4 =
        ((size_t)b * QKV_ + (size_t)(HQ_ + HKV_) * DH_ + (size_t)h * DH_) >> 2;
    for (int t = 0; t < TILES_; ++t) {
      const int pbase = cbase + t * 16;
      __syncthreads();
      if (pbase + 15 < cur) {
        stage_async(gK + (size_t)t * 16 * DH_, gV + (size_t)t * 16 * DH_,
                    sK[0], sV[0], lane);
        asm volatile("s_wait_asynccnt 0x0" ::: "memory");
      } else {
        for (int i = lane; i < 512; i += 32) {
          const int row = i >> 5, c4 = i & 31;
          const int p = pbase + row;
          float4 kv = make_float4(0.f, 0.f, 0.f, 0.f);
          float4 vv = make_float4(0.f, 0.f, 0.f, 0.f);
          if (p < cur) {
            const size_t base4 = ((size_t)(b * HKV_ + h) * S_ + p) * (DH_ / 4);
            kv = K4[base4 + c4];
            vv = V4[base4 + c4];
          } else if (p == cur) {
            kv = Kn4[knbase4 + c4];
            vv = Qk4[vnbase4 + c4];
          }
          *reinterpret_cast<float4*>(&sK[0][row * LDP + c4 * 4]) = kv;
          *reinterpret_cast<float4*>(&sV[0][row * LDP + c4 * 4]) = vv;
        }
      }
      __syncthreads();
      compute_tile(sK[0], sV[0], pbase);
    }
  }

  // Partials: 4x128 out + 4 m + 4 l per (b, h, chunk).
  if (hi == 0) {
#pragma unroll
    for (int nt = 0; nt < 8; ++nt)
#pragma unroll
      for (int g = 0; g < G_; ++g) P[g * DH_ + nt * 16 + nrow] = acc[nt][g];
  }
  if (lane == 0) {
#pragma unroll
    for (int g = 0; g < G_; ++g) {
      P[G_ * DH_ + g] = mr[g];
      P[G_ * DH_ + G_ + g] = lr[g];
    }
  }
}

// ---------------------------------------------------------------------------
// Kernel 4: combine chunk partials -> attn heads (B x HQ x DH).
// ---------------------------------------------------------------------------
__global__ void k_combine(const float* __restrict__ part, float* __restrict__ attn) {
  const int d = threadIdx.x;      // 0..127
  const int hq = blockIdx.x;      // 0..31
  const int b = blockIdx.y;
  const int h = hq / G_, g = hq % G_;
  const int stride = G_ * DH_ + 2 * G_;
  const float* base = part + ((size_t)(b * HKV_ + h)) * NCHUNK_ * stride;
  float M = -1e30f;
  for (int c = 0; c < NCHUNK_; ++c) M = fmaxf(M, base[c * stride + G_ * DH_ + g]);
  float L = 0.f, o = 0.f;
  for (int c = 0; c < NCHUNK_; ++c) {
    const float* pc = base + c * stride;
    const float w = __expf(pc[G_ * DH_ + g] - M);
    L += w * pc[G_ * DH_ + G_ + g];
    o += w * pc[g * DH_ + d];
  }
  attn[((size_t)b * HQ_ + hq) * DH_ + d] = o / L;
}

// ---------------------------------------------------------------------------
// Kernel 5: dense = attn(4x4096) @ wo(4096x4096) -> d_out.
// Same dual-tile b64-load scheme as k_qkv.
// ---------------------------------------------------------------------------
__global__ void k_wo(const float* __restrict__ attn, const float* __restrict__ wo,
                     float* __restrict__ out) {
  const int lane = threadIdx.x;
  const int nrow = lane & 15, hi = lane >> 4;
  const int n0 = blockIdx.x * 32;
  const int mrow = nrow & 3;
  const float2* a2 = reinterpret_cast<const float2*>(attn);
  const float2* w2 = reinterpret_cast<const float2*>(wo);
  v8f acc0 = {}, acc1 = {};
  for (int k0 = 0; k0 < D_; k0 += 4) {
    const int ka = k0 + 2 * hi;
    const float2 av = a2[((size_t)mrow * D_ + ka) >> 1];
    const float2 w0 = w2[((size_t)ka * D_ + n0 + 2 * nrow) >> 1];
    const float2 w1 = w2[((size_t)(ka + 1) * D_ + n0 + 2 * nrow) >> 1];
    v2f a;  a.x = av.x;  a.y = av.y;
    v2f b0; b0.x = w0.x; b0.y = w1.x;
    v2f b1; b1.x = w0.y; b1.y = w1.y;
    acc0 = wmma4(a, b0, acc0);
    acc1 = wmma4(a, b1, acc1);
  }
#pragma unroll
  for (int r = 0; r < 8; ++r) {
    const int m = r + 8 * hi;
    if (m < B_) {
      float2 st; st.x = acc0[r]; st.y = acc1[r];
      *reinterpret_cast<float2*>(&out[(size_t)m * D_ + n0 + 2 * nrow]) = st;
    }
  }
}

// ---------------------------------------------------------------------------
extern "C" void kernel_launch(void* const* d_in, const int* in_sizes, int n_in,
                              void* d_out, int out_size, void* d_ws, size_t ws_size,
                              hipStream_t stream) {
  (void)in_sizes; (void)n_in; (void)out_size; (void)ws_size;
  const float* x    = (const float*)d_in[0];
  const float* wqkv = (const float*)d_in[1];
  const float* wo   = (const float*)d_in[2];
  const float* rot  = (const float*)d_in[3];
  const float* ck   = (const float*)d_in[4];
  const float* cv   = (const float*)d_in[5];
  const int*   cur  = (const int*)d_in[6];

  float* ws   = (float*)d_ws;
  float* qkv  = ws;                                   // B*QKV        = 24576
  float* qrot = qkv  + (size_t)B_ * QKV_;             // B*HQ*DH      = 16384
  float* krot = qrot + (size_t)B_ * HQ_ * DH_;        // B*HKV*DH     = 4096
  float* attn = krot + (size_t)B_ * HKV_ * DH_;       // B*HQ*DH      = 16384
  float* part = attn + (size_t)B_ * HQ_ * DH_;        // 4*8*32*520   = 532480

  k_qkv    <<<QKV_ / 32,               32,  0, stream>>>(x, wqkv, qkv);
  k_rope   <<<B_ * 3 * 8,              32,  0, stream>>>(qkv, rot, qrot, krot);
  k_sdpa   <<<dim3(NCHUNK_, HKV_, B_), 32,  0, stream>>>(qrot, krot, qkv, ck, cv, cur, part);
  k_combine<<<dim3(HQ_, B_),           DH_, 0, stream>>>(part, attn);
  k_wo     <<<D_ / 32,                 32,  0, stream>>>(attn, wo, (float*)d_out);
}